// GAT_27358941676116
// MI455X (gfx1250) — compile-verified
//
#include <hip/hip_runtime.h>
#include <math.h>

typedef __attribute__((ext_vector_type(2))) float v2f;
typedef __attribute__((ext_vector_type(8))) float v8f;

#define GAT_N      10000
#define GAT_E      320000
#define GAT_IN     256
#define GAT_HID    64
#define GAT_OUT    64
#define GAT_H1     8
#define GAT_H2     1
#define NEG_SLOPE  0.2f

// ---------------------------------------------------------------- fill
__global__ void k_fill(float* __restrict__ p, float v, int n) {
    int i = blockIdx.x * blockDim.x + threadIdx.x;
    if (i < n) p[i] = v;
}

// ------------------------------------------------- GEMM: C = A(MxK) @ B(KxN)
// Full-FP32 WMMA (V_WMMA_F32_16X16X4_F32). One wave owns one 16x16 C tile.
// A 16x4 layout: lanes 0-15 hold {K=0,K=1}, lanes 16-31 hold {K=2,K=3}, M=lane&15.
// B 4x16 layout mirrors A (row striped across lanes within a VGPR).
// C/D: VGPR i -> M = i + 8*(lane>=16), N = lane&15.
__global__ void k_gemm_wmma(const float* __restrict__ A,
                            const float* __restrict__ B,
                            float* __restrict__ C,
                            int M, int K, int N) {
    const int lane   = threadIdx.x & 31;
    const int wave   = threadIdx.x >> 5;
    const int nTiles = N >> 4;
    const int nTile  = blockIdx.y * (blockDim.x >> 5) + wave;
    if (nTile >= nTiles) return;            // wave-uniform: EXEC stays all-ones
    const int mBase = blockIdx.x << 4;
    const int half  = lane >> 4;            // 0 for lanes 0-15, 1 for 16-31
    const int l15   = lane & 15;
    const int mRow  = mBase + l15;
    const int nCol  = (nTile << 4) + l15;
    const int kOff  = half << 1;            // 0 or 2

    const float* aRow = A + (size_t)mRow * K;
    v8f acc = {};
    for (int k = 0; k < K; k += 4) {
        v2f a, b;
        a.x = aRow[k + kOff];
        a.y = aRow[k + kOff + 1];
        b.x = B[(size_t)(k + kOff)     * N + nCol];
        b.y = B[(size_t)(k + kOff + 1) * N + nCol];
        acc = __builtin_amdgcn_wmma_f32_16x16x4_f32(
            /*neg_a=*/false, a, /*neg_b=*/false, b,
            /*c_mod=*/(short)0, acc, /*reuse_a=*/false, /*reuse_b=*/false);
    }
    const int rBase = mBase + half * 8;
#pragma unroll
    for (int i = 0; i < 8; ++i)
        C[(size_t)(rBase + i) * N + nCol] = acc[i];
}

// ---------------------------------------- per-(node,head) attention scores
__global__ void k_attn_scores(const float* __restrict__ ft,
                              const float* __restrict__ al,
                              const float* __restrict__ ar,
                              float* __restrict__ el,
                              float* __restrict__ er,
                              int N, int H, int D) {
    int idx = blockIdx.x * blockDim.x + threadIdx.x;
    if (idx >= N * H) return;
    int h = idx % H;
    const float* f   = ft + (size_t)idx * D;       // ft is [N, H, D] contiguous
    const float* alh = al + (size_t)h * D;
    const float* arh = ar + (size_t)h * D;
    float sl = 0.f, sr = 0.f;
    for (int d = 0; d < D; ++d) {
        float v = f[d];
        sl += v * alh[d];
        sr += v * arh[d];
    }
    el[idx] = sl;
    er[idx] = sr;
}

// ------------------------------------------------- float atomic max (CAS)
__device__ __forceinline__ void atomicMaxF(float* addr, float val) {
    unsigned int* ua  = (unsigned int*)addr;
    unsigned int  old = *ua;
    while (__uint_as_float(old) < val) {
        unsigned int prev = atomicCAS(ua, old, __float_as_uint(val));
        if (prev == old) break;
        old = prev;
    }
}

// ------------------------------- pass A: e = leaky_relu(el[src]+er[dst]); m = segmax
__global__ void k_edge_max(const int* __restrict__ src, const int* __restrict__ dst,
                           const float* __restrict__ el, const float* __restrict__ er,
                           float* __restrict__ e, float* __restrict__ m,
                           int E, int H) {
    int idx = blockIdx.x * blockDim.x + threadIdx.x;
    if (idx >= E * H) return;
    int ed = idx / H, h = idx % H;
    int s = src[ed], t = dst[ed];
    float v = el[s * H + h] + er[t * H + h];
    v = (v > 0.f) ? v : NEG_SLOPE * v;
    e[idx] = v;
    atomicMaxF(&m[t * H + h], v);
}

// ------------------------------- pass B: w = exp(e - m[dst]); denom = segsum(w)
__global__ void k_edge_expsum(const int* __restrict__ dst,
                              const float* __restrict__ m,
                              float* __restrict__ e,
                              float* __restrict__ denom,
                              int E, int H) {
    int idx = blockIdx.x * blockDim.x + threadIdx.x;
    if (idx >= E * H) return;
    int ed = idx / H, h = idx % H;
    int t  = dst[ed];
    float w = expf(e[idx] - m[t * H + h]);
    e[idx] = w;
    atomicAdd(&denom[t * H + h], w);
}

// ------- pass C: out[dst,h,d] += ft[src,h,d] * (w / denom[dst,h]); coalesced over d
__global__ void k_edge_aggregate(const int* __restrict__ src, const int* __restrict__ dst,
                                 const float* __restrict__ w, const float* __restrict__ denom,
                                 const float* __restrict__ ft, float* __restrict__ out,
                                 int E, int H, int D) {
    int HD  = H * D;
    long long total = (long long)E * HD;
    long long idx = (long long)blockIdx.x * blockDim.x + threadIdx.x;
    if (idx >= total) return;
    int ed = (int)(idx / HD);
    int r  = (int)(idx % HD);
    int h  = r / D;
    int s = src[ed], t = dst[ed];
    float alpha = w[(size_t)ed * H + h] / denom[(size_t)t * H + h];
    atomicAdd(&out[(size_t)t * HD + r], ft[(size_t)s * HD + r] * alpha);
}

// ------------------------------------------------- ELU(x + bias), in place
__global__ void k_elu_bias(float* __restrict__ x, const float* __restrict__ b,
                           int n, int cols) {
    int i = blockIdx.x * blockDim.x + threadIdx.x;
    if (i >= n) return;
    float v = x[i] + b[i % cols];
    x[i] = (v > 0.f) ? v : expm1f(v);
}

// ------------------------------------------------- final: out = agg2 + b2
__global__ void k_bias_out(const float* __restrict__ a, const float* __restrict__ b,
                           float* __restrict__ o, int n, int cols) {
    int i = blockIdx.x * blockDim.x + threadIdx.x;
    if (i >= n) return;
    o[i] = a[i] + b[i % cols];
}

static inline int cdiv(long long a, int b) { return (int)((a + b - 1) / b); }

extern "C" void kernel_launch(void* const* d_in, const int* in_sizes, int n_in,
                              void* d_out, int out_size, void* d_ws, size_t ws_size,
                              hipStream_t stream) {
    (void)in_sizes; (void)n_in; (void)out_size; (void)ws_size;
    // inputs (setup_inputs order)
    const float* x   = (const float*)d_in[0];
    const int*   src = (const int*)  d_in[1];
    const int*   dst = (const int*)  d_in[2];
    const float* W1  = (const float*)d_in[3];
    const float* al1 = (const float*)d_in[4];
    const float* ar1 = (const float*)d_in[5];
    const float* b1  = (const float*)d_in[6];
    const float* W2  = (const float*)d_in[7];
    const float* al2 = (const float*)d_in[8];
    const float* ar2 = (const float*)d_in[9];
    const float* b2  = (const float*)d_in[10];
    float* out = (float*)d_out;

    const int N = GAT_N, E = GAT_E;
    const int F1 = GAT_H1 * GAT_HID;   // 512
    const int F2 = GAT_H2 * GAT_OUT;   // 64

    // workspace carve-up (floats)
    float* ws = (float*)d_ws;
    float* ft1    = ws;                ws += (size_t)N * F1;        // 5.12M
    float* agg1   = ws;                ws += (size_t)N * F1;        // 5.12M
    float* ft2    = ws;                ws += (size_t)N * F2;
    float* agg2   = ws;                ws += (size_t)N * F2;
    float* el1    = ws;                ws += (size_t)N * GAT_H1;
    float* er1    = ws;                ws += (size_t)N * GAT_H1;
    float* m1     = ws;                ws += (size_t)N * GAT_H1;
    float* den1   = ws;                ws += (size_t)N * GAT_H1;
    float* el2    = ws;                ws += (size_t)N * GAT_H2;
    float* er2    = ws;                ws += (size_t)N * GAT_H2;
    float* m2     = ws;                ws += (size_t)N * GAT_H2;
    float* den2   = ws;                ws += (size_t)N * GAT_H2;
    float* ebuf1  = ws;                ws += (size_t)E * GAT_H1;    // 2.56M
    float* ebuf2  = ws;                ws += (size_t)E * GAT_H2;

    const int T = 256;
    const float NEG_INF = -__builtin_huge_valf();

    // ---- init accumulators (workspace is poisoned; must re-init every call)
    k_fill<<<cdiv((long long)N * F1, T), T, 0, stream>>>(agg1, 0.f, N * F1);
    k_fill<<<cdiv((long long)N * GAT_H1, T), T, 0, stream>>>(m1,   NEG_INF, N * GAT_H1);
    k_fill<<<cdiv((long long)N * GAT_H1, T), T, 0, stream>>>(den1, 0.f,     N * GAT_H1);
    k_fill<<<cdiv((long long)N * F2, T), T, 0, stream>>>(agg2, 0.f, N * F2);
    k_fill<<<cdiv((long long)N * GAT_H2, T), T, 0, stream>>>(m2,   NEG_INF, N * GAT_H2);
    k_fill<<<cdiv((long long)N * GAT_H2, T), T, 0, stream>>>(den2, 0.f,     N * GAT_H2);

    // ---- layer 1: ft1 = x @ W1   (M=10000, K=256, N=512)
    {
        dim3 grid(N / 16, (F1 / 16 + 7) / 8);
        k_gemm_wmma<<<grid, 256, 0, stream>>>(x, W1, ft1, N, GAT_IN, F1);
    }
    k_attn_scores<<<cdiv((long long)N * GAT_H1, T), T, 0, stream>>>(
        ft1, al1, ar1, el1, er1, N, GAT_H1, GAT_HID);
    k_edge_max<<<cdiv((long long)E * GAT_H1, T), T, 0, stream>>>(
        src, dst, el1, er1, ebuf1, m1, E, GAT_H1);
    k_edge_expsum<<<cdiv((long long)E * GAT_H1, T), T, 0, stream>>>(
        dst, m1, ebuf1, den1, E, GAT_H1);
    k_edge_aggregate<<<cdiv((long long)E * F1, T), T, 0, stream>>>(
        src, dst, ebuf1, den1, ft1, agg1, E, GAT_H1, GAT_HID);
    k_elu_bias<<<cdiv((long long)N * F1, T), T, 0, stream>>>(agg1, b1, N * F1, F1);

    // ---- layer 2: ft2 = h @ W2   (M=10000, K=512, N=64)
    {
        dim3 grid(N / 16, (F2 / 16 + 7) / 8);
        k_gemm_wmma<<<grid, 256, 0, stream>>>(agg1, W2, ft2, N, F1, F2);
    }
    k_attn_scores<<<cdiv((long long)N * GAT_H2, T), T, 0, stream>>>(
        ft2, al2, ar2, el2, er2, N, GAT_H2, GAT_OUT);
    k_edge_max<<<cdiv((long long)E * GAT_H2, T), T, 0, stream>>>(
        src, dst, el2, er2, ebuf2, m2, E, GAT_H2);
    k_edge_expsum<<<cdiv((long long)E * GAT_H2, T), T, 0, stream>>>(
        dst, m2, ebuf2, den2, E, GAT_H2);
    k_edge_aggregate<<<cdiv((long long)E * F2, T), T, 0, stream>>>(
        src, dst, ebuf2, den2, ft2, agg2, E, GAT_H2, GAT_OUT);

    // ---- mean over heads (H2=1) + bias
    k_bias_out<<<cdiv((long long)N * F2, T), T, 0, stream>>>(agg2, b2, out, N * F2, F2);
}